// mem_update_27092653703300
// MI455X (gfx1250) — compile-verified
//
#include <hip/hip_runtime.h>

// LIF membrane scan, T=4 (reference shape [4, 32, 256, 32, 32] f32):
//   mem_t = (mem_{t-1} - spike_{t-1} * 0.5) * 0.25 + x_t
//   spike_t = round(clamp(mem_t, 0, 1))   (round-half-to-even == jnp.round)
//
// Memory-bound streaming op: 134 MB in + 134 MB out => ~11.5 us floor at
// 23.3 TB/s. No matmul structure -> WMMA inapplicable. Data path uses the
// CDNA5 async global->LDS engine (GLOBAL_LOAD_ASYNC_TO_LDS_B128, ASYNCcnt,
// s_wait_asynccnt; CDNA5 ISA): per-lane 16B copies into a
// conflict-free LDS tile, staged waits (async completions are in-order) so
// timestep-t compute overlaps the remaining copies, then NT b128 stores
// straight to global (stream > 192 MB L2, zero reuse).

typedef __attribute__((ext_vector_type(4))) float v4f;
typedef int v4i __attribute__((vector_size(16)));  // matches builtin param type

#ifndef __has_builtin
#define __has_builtin(x) 0
#endif

#if __has_builtin(__builtin_amdgcn_global_load_async_to_lds_b128) && \
    __has_builtin(__builtin_amdgcn_s_wait_asynccnt)
#define ASYNC_BUILTIN 1
#else
#define ASYNC_BUILTIN 0
#endif

typedef __attribute__((address_space(1))) v4i* gbl_v4i_p;  // global (AS1)
typedef __attribute__((address_space(3))) v4i* lds_v4i_p;  // LDS (AS3)

__device__ __forceinline__ float quant1(float m) {
    float c = __builtin_fminf(__builtin_fmaxf(m, 0.0f), 1.0f);  // fuses to v_max ... clamp
    return __builtin_rintf(c);                                   // v_rndne_f32 (half-to-even)
}

__device__ __forceinline__ v4f quant1v(v4f m) {
    v4f r;
    r.x = quant1(m.x);
    r.y = quant1(m.y);
    r.z = quant1(m.z);
    r.w = quant1(m.w);
    return r;
}

// ---- async copy + wait primitives --------------------------------------

__device__ __forceinline__ void async_copy16(const v4f* gsrc, v4f* lds_generic) {
#if ASYNC_BUILTIN
    __builtin_amdgcn_global_load_async_to_lds_b128(
        (gbl_v4i_p)gsrc, (lds_v4i_p)lds_generic, 0, 0);
#else
    unsigned long long ga = (unsigned long long)gsrc;
    unsigned lo = (unsigned)(unsigned long long)(lds_v4i_p)lds_generic;
    asm volatile("global_load_async_to_lds_b128 %0, %1, off"
                 :: "v"(lo), "v"(ga) : "memory");
#endif
}

__device__ __forceinline__ void wait_async_le(int n) {
#if ASYNC_BUILTIN
    switch (n) {
        case 3: __builtin_amdgcn_s_wait_asynccnt(3); break;
        case 2: __builtin_amdgcn_s_wait_asynccnt(2); break;
        case 1: __builtin_amdgcn_s_wait_asynccnt(1); break;
        default: __builtin_amdgcn_s_wait_asynccnt(0); break;
    }
#else
    switch (n) {
        case 3: asm volatile("s_wait_asynccnt 0x3" ::: "memory"); break;
        case 2: asm volatile("s_wait_asynccnt 0x2" ::: "memory"); break;
        case 1: asm volatile("s_wait_asynccnt 0x1" ::: "memory"); break;
        default: asm volatile("s_wait_asynccnt 0x0" ::: "memory"); break;
    }
#endif
    asm volatile("" ::: "memory");  // keep LDS reads from drifting above the wait
}

// ---- main kernel -------------------------------------------------------

__global__ __launch_bounds__(256) void lif_scan_v4_async(
    const float* __restrict__ x, float* __restrict__ out,
    long long nvec, long long svec)  // svec: per-timestep stride in float4 units
{
    __shared__ v4f tile[4][256];  // 16 KB; lanes hit consecutive 16B -> conflict-free b128
    const int tid = threadIdx.x;
    const long long i = (long long)blockIdx.x * blockDim.x + tid;
    if (i >= nvec) return;  // partial last block: inactive lanes issue nothing (EXEC)

    const v4f* __restrict__ xv = (const v4f*)x;
    v4f*       __restrict__ ov = (v4f*)out;

    // Issue all 4 timestep copies up front; only the arithmetic is serial.
    async_copy16(xv + 0 * svec + i, &tile[0][tid]);
    async_copy16(xv + 1 * svec + i, &tile[1][tid]);
    async_copy16(xv + 2 * svec + i, &tile[2][tid]);
    async_copy16(xv + 3 * svec + i, &tile[3][tid]);

    // Async completions are in-order (ISA 08 §4.1): stage waits, overlap compute.
    wait_async_le(3);
    v4f mem = tile[0][tid];        // t=0: mem_old=0, spike=0 -> mem = x0
    v4f s   = quant1v(mem);
    __builtin_nontemporal_store(s, ov + 0 * svec + i);

    wait_async_le(2);
    mem = (mem - s * 0.5f) * 0.25f + tile[1][tid];
    s   = quant1v(mem);
    __builtin_nontemporal_store(s, ov + 1 * svec + i);

    wait_async_le(1);
    mem = (mem - s * 0.5f) * 0.25f + tile[2][tid];
    s   = quant1v(mem);
    __builtin_nontemporal_store(s, ov + 2 * svec + i);

    wait_async_le(0);
    mem = (mem - s * 0.5f) * 0.25f + tile[3][tid];
    s   = quant1v(mem);
    __builtin_nontemporal_store(s, ov + 3 * svec + i);
}

// Scalar tail for S not divisible by 4 (not hit at the reference shape).
__global__ __launch_bounds__(256) void lif_scan_tail(
    const float* __restrict__ x, float* __restrict__ out,
    long long start, long long S)
{
    long long i = start + (long long)blockIdx.x * blockDim.x + threadIdx.x;
    if (i >= S) return;

    float mem = 0.0f, spike = 0.0f;
#pragma unroll
    for (int t = 0; t < 4; ++t) {
        mem   = (mem - spike * 0.5f) * 0.25f + x[(long long)t * S + i];
        spike = quant1(mem);
        out[(long long)t * S + i] = spike;
    }
}

extern "C" void kernel_launch(void* const* d_in, const int* in_sizes, int n_in,
                              void* d_out, int out_size, void* d_ws, size_t ws_size,
                              hipStream_t stream) {
    (void)n_in; (void)out_size; (void)d_ws; (void)ws_size;

    const float* x   = (const float*)d_in[0];
    float*       out = (float*)d_out;

    const long long total = (long long)in_sizes[0];  // T * B * C * H * W
    const long long S     = total / 4;               // per-timestep elements (T = 4)
    const long long nvec  = S / 4;                   // full float4 columns
    const long long rem   = S - nvec * 4;

    const int block = 256;  // 8 wave32 waves per block
    if (nvec > 0) {
        const long long grid = (nvec + block - 1) / block;
        lif_scan_v4_async<<<(unsigned int)grid, block, 0, stream>>>(x, out, nvec, nvec);
    }
    if (rem > 0) {
        const long long grid = (rem + block - 1) / block;
        lif_scan_tail<<<(unsigned int)grid, block, 0, stream>>>(x, out, nvec * 4, S);
    }
}